// TVLoss_5016521802106
// MI455X (gfx1250) — compile-verified
//
#include <hip/hip_runtime.h>

#define R_DIM      256
#define R_SQ       (R_DIM * R_DIM)
#define LAMBDA_TV  1e-05f
#define EPS_TV     1e-09f

// ---------------------------------------------------------------------------
// Per-cell TV gradient with scatter-add. (Placed first so the disasm snippet
// shows this hot kernel.)
//
// cells are consecutive linear indices into the 256^3 grid, so the four link
// loads (cell, cell+1, cell+R, cell+R^2) are fully coalesced across lanes.
// Density gathers are random 4B loads; contributions scatter via native
// global_atomic_add_f32 (relaxed, device scope — no CAS loop).
// ---------------------------------------------------------------------------
__global__ __launch_bounds__(256) void tv_grad_kernel(
    const float* __restrict__ density,   // (N,1) f32
    const int*   __restrict__ links,     // (R,R,R) i32, flat
    const int*   __restrict__ cells,     // (sparse_num,) i32
    float*       __restrict__ grad,      // (N,1) f32 out (pre-zeroed)
    int num_cells)
{
    int i = blockIdx.x * blockDim.x + threadIdx.x;
    if (i >= num_cells) return;

    int cell = cells[i];
    // R == 256: decompose with shifts/masks (z fastest-varying).
    int z = cell & (R_DIM - 1);
    int y = (cell >> 8) & (R_DIM - 1);
    int x = cell >> 16;

    bool in100 = (x + 1) < R_DIM;
    bool in010 = (y + 1) < R_DIM;
    bool in001 = (z + 1) < R_DIM;

    // gfx1250: warm L2 for the strided neighbor cachelines (+R, +R^2) while
    // the base line is fetched. Predicated per-lane; emits global_prefetch_b8.
    if (in010) __builtin_prefetch(links + cell + R_DIM, 0, 0);
    if (in100) __builtin_prefetch(links + cell + R_SQ,  0, 0);

    int l000 = links[cell];
    int l001 = in001 ? links[cell + 1]     : -1;
    int l010 = in010 ? links[cell + R_DIM] : -1;
    int l100 = in100 ? links[cell + R_SQ]  : -1;

    float v000 = (l000 >= 0) ? density[l000] : 0.f;
    float v100 = in100 ? ((l100 >= 0) ? density[l100] : 0.f) : v000;
    float v010 = in010 ? ((l010 >= 0) ? density[l010] : 0.f) : v000;
    float v001 = in001 ? ((l001 >= 0) ? density[l001] : 0.f) : v000;

    const float s = R_DIM * 0.5f;  // 128.0
    float dx = (v100 - v000) * s;
    float dy = (v010 - v000) * s;
    float dz = (v001 - v000) * s;

    float idelta = LAMBDA_TV * rsqrtf(EPS_TV + dx * dx + dy * dy + dz * dz);

    // Scatter-add; index -1 is the reference's discard row -> skip.
    // Relaxed + agent scope => native non-returning global_atomic_add_f32.
    float c000 = -(dx + dy + dz) * idelta;
    if (l000 >= 0) __hip_atomic_fetch_add(grad + l000, c000,
                                          __ATOMIC_RELAXED, __HIP_MEMORY_SCOPE_AGENT);
    if (l100 >= 0) __hip_atomic_fetch_add(grad + l100, dx * idelta,
                                          __ATOMIC_RELAXED, __HIP_MEMORY_SCOPE_AGENT);
    if (l010 >= 0) __hip_atomic_fetch_add(grad + l010, dy * idelta,
                                          __ATOMIC_RELAXED, __HIP_MEMORY_SCOPE_AGENT);
    if (l001 >= 0) __hip_atomic_fetch_add(grad + l001, dz * idelta,
                                          __ATOMIC_RELAXED, __HIP_MEMORY_SCOPE_AGENT);
}

// ---------------------------------------------------------------------------
// Zero the 32 MB gradient buffer (harness poisons d_out with 0xAA).
// float4 stores -> global_store_b128; HBM-limited (~1.4 us at 23.3 TB/s).
// ---------------------------------------------------------------------------
__global__ __launch_bounds__(256) void tv_zero_kernel(float* __restrict__ out, int n) {
    int i = (blockIdx.x * blockDim.x + threadIdx.x) * 4;
    if (i + 3 < n) {
        float4 z4 = make_float4(0.f, 0.f, 0.f, 0.f);
        *reinterpret_cast<float4*>(out + i) = z4;
    } else {
        // Only the last (partial) quad lands here: <= 3 scalar stores.
        while (i < n) { out[i] = 0.f; ++i; }
    }
}

extern "C" void kernel_launch(void* const* d_in, const int* in_sizes, int n_in,
                              void* d_out, int out_size, void* d_ws, size_t ws_size,
                              hipStream_t stream) {
    (void)n_in; (void)d_ws; (void)ws_size;

    const float* density = (const float*)d_in[0];   // (N*C,) f32
    const int*   links   = (const int*)d_in[1];     // (R^3,) i32
    const int*   cells   = (const int*)d_in[2];     // (sparse_num,) i32
    float*       grad    = (float*)d_out;           // (N*C,) f32

    int num_cells = in_sizes[2];

    // 1) Zero the output (poisoned by harness; grad is zeros + sparse adds).
    {
        int n4_threads = (out_size + 3) / 4;
        int blocks = (n4_threads + 255) / 256;
        tv_zero_kernel<<<blocks, 256, 0, stream>>>(grad, out_size);
    }

    // 2) Sparse TV gradient scatter. 256 threads = 8 wave32 waves per block.
    {
        int blocks = (num_cells + 255) / 256;
        tv_grad_kernel<<<blocks, 256, 0, stream>>>(density, links, cells, grad, num_cells);
    }
}